// GCNModel_62508954025989
// MI455X (gfx1250) — compile-verified
//
#include <hip/hip_runtime.h>
#include <math.h>

typedef __attribute__((ext_vector_type(2))) float v2f;
typedef __attribute__((ext_vector_type(8))) float v8f;

// ---------------------------------------------------------------------------
// x0 = concat(user_emb, item_emb); out (ui accumulator) = x0; het = 3*x0
// ---------------------------------------------------------------------------
__global__ void k_init(const float* __restrict__ ue, const float* __restrict__ ie,
                       float* __restrict__ x0, float* __restrict__ out,
                       float* __restrict__ het, long long nU64, long long nTot64)
{
    long long i = (long long)blockIdx.x * blockDim.x + threadIdx.x;
    if (i >= nTot64) return;
    float v = (i < nU64) ? ue[i] : ie[i - nU64];
    x0[i]  = v;
    out[i] = v;
    het[i] = 3.0f * v;
}

// ---------------------------------------------------------------------------
// degree histogram (u32 atomics) and in-place deg -> rsqrt(max(deg,1))
// ---------------------------------------------------------------------------
__global__ void k_hist(const int* __restrict__ idx, unsigned* __restrict__ deg, int nE)
{
    int e = blockIdx.x * blockDim.x + threadIdx.x;
    if (e < nE) atomicAdd(&deg[idx[e]], 1u);
}

__global__ void k_deg2rnorm(float* __restrict__ buf, int n)
{
    int i = blockIdx.x * blockDim.x + threadIdx.x;
    if (i >= n) return;
    unsigned u = __float_as_uint(buf[i]);     // buffer holds u32 counts
    buf[i] = rsqrtf(fmaxf((float)u, 1.0f));
}

// ---------------------------------------------------------------------------
// Edge SpMM: one wave per edge. Wave-uniform src/dst (readfirstlane) so index
// loads scalarize; lane l handles features 2l,2l+1: gather float2 from
// x[src]*dnorm_src[src], scatter via global_atomic_add_f32 into agg[dst].
// Both x and agg (76.8MB each) fit in the 192MB L2 -> L2-resident RMW.
// ---------------------------------------------------------------------------
__global__ void k_edge(const int* __restrict__ src, const int* __restrict__ dst,
                       const float* __restrict__ x, const float* __restrict__ dns,
                       float* __restrict__ agg, int nE)
{
    const int lane = threadIdx.x & 31;
    long long w = (long long)blockIdx.x * (blockDim.x >> 5) + (threadIdx.x >> 5);
    if (w >= nE) return;
    int s = __builtin_amdgcn_readfirstlane(src[w]);
    int d = __builtin_amdgcn_readfirstlane(dst[w]);
    float dn = dns[s];
    const float2 v = *reinterpret_cast<const float2*>(x + (size_t)s * 64 + lane * 2);
    float* p = agg + (size_t)d * 64 + lane * 2;
    atomicAdd(p,     v.x * dn);
    atomicAdd(p + 1, v.y * dn);
}

// ---------------------------------------------------------------------------
// Post-layer node kernel: one wave per 16 nodes (4 waves / 128-thread block).
//  1) coalesced read agg, scale by dnorm_dst, LeakyReLU(0.5), write next-x
//     (in place over agg) and stage the 16x64 tile in LDS (row stride 68 words
//     -> bank = (4m+k)%64, conflict-free for the WMMA-layout b64 reads).
//  2) 16x V_WMMA_F32_16X16X4_F32 with A=B=X chunk -> C = X·X^T (16x16 Gram).
//     f32 16x4 A layout (lanes0-15: M, K0/K1; lanes16-31: M, K2/K3) is
//     register-identical to the 4x16 B layout of X^T, so one v2f feeds both.
//  3) diagonal of C = row sums of squares -> inv = 1/max(sqrt(sq),1e-12)
//  4) acc[row] += x * inv (coalesced RMW, x kept in registers from step 1).
// ---------------------------------------------------------------------------
#define ROWSTRIDE 68               // words; multiple of 4 (16B LDS alignment)
#define WSEG (16 * ROWSTRIDE + 16) // per-wave LDS floats (tile + 16 sq slots)

__global__ void k_post(const float* __restrict__ agg, const float* __restrict__ dnd,
                       float* __restrict__ xnext, float* __restrict__ acc, int nNodes)
{
    __shared__ float lds[4 * WSEG];
    const int lane = threadIdx.x & 31;
    const int wv   = threadIdx.x >> 5;
    float* L  = lds + wv * WSEG;
    float* SQ = L + 16 * ROWSTRIDE;
    const int base = (blockIdx.x * 4 + wv) * 16;
    if (base >= nNodes) return;                 // wave-uniform; EXEC stays full
    const size_t g = (size_t)base * 64;

    float4 xr[8];
#pragma unroll
    for (int i = 0; i < 8; ++i) {
        int flat = i * 128 + lane * 4;          // lanes cover 2 rows / iter
        int row  = flat >> 6;
        int col  = flat & 63;
        float dn = dnd[base + row];
        float4 a = *reinterpret_cast<const float4*>(agg + g + flat);
        float4 x;
        x.x = a.x * dn; x.y = a.y * dn; x.z = a.z * dn; x.w = a.w * dn;
        x.x = (x.x >= 0.0f) ? x.x : 0.5f * x.x; // LeakyReLU(0.5)
        x.y = (x.y >= 0.0f) ? x.y : 0.5f * x.y;
        x.z = (x.z >= 0.0f) ? x.z : 0.5f * x.z;
        x.w = (x.w >= 0.0f) ? x.w : 0.5f * x.w;
        xr[i] = x;
        *reinterpret_cast<float4*>(&L[row * ROWSTRIDE + col]) = x;
        *reinterpret_cast<float4*>(xnext + g + flat) = x;   // xnext may alias agg
    }

    // Gram matrix via WMMA f32 16x16x4, accumulating over K=64 in chunks of 4
    v8f c = {};
    const int m  = lane & 15;
    const int kh = (lane >> 4) << 1;            // 0 for lanes 0-15, 2 for 16-31
#pragma unroll
    for (int kc = 0; kc < 16; ++kc) {
        v2f a = *reinterpret_cast<const v2f*>(&L[m * ROWSTRIDE + kc * 4 + kh]);
        c = __builtin_amdgcn_wmma_f32_16x16x4_f32(false, a, false, a,
                                                  (short)0, c, false, false);
    }

    // diagonal: C[r][r] at (vgpr r, lane r); C[8+r][8+r] at (vgpr r, lane 24+r)
#pragma unroll
    for (int r = 0; r < 8; ++r) {
        float dv = c[r];
        if (lane == r)      SQ[r]     = dv;
        if (lane == 24 + r) SQ[8 + r] = dv;
    }

#pragma unroll
    for (int i = 0; i < 8; ++i) {
        int flat = i * 128 + lane * 4;
        int row  = flat >> 6;
        float inv = 1.0f / fmaxf(sqrtf(SQ[row]), 1e-12f);
        if (base + row < nNodes) {
            float4 av = *reinterpret_cast<float4*>(acc + g + flat);
            float4 x  = xr[i];
            av.x += x.x * inv; av.y += x.y * inv;
            av.z += x.z * inv; av.w += x.w * inv;
            *reinterpret_cast<float4*>(acc + g + flat) = av;
        }
    }
}

// ---------------------------------------------------------------------------
// out += LeakyReLU(het, 0.5)
// ---------------------------------------------------------------------------
__global__ void k_final(float* __restrict__ out, const float* __restrict__ het,
                        long long n)
{
    long long i = (long long)blockIdx.x * blockDim.x + threadIdx.x;
    if (i >= n) return;
    float h = het[i];
    h = (h >= 0.0f) ? h : 0.5f * h;
    out[i] += h;
}

// ---------------------------------------------------------------------------
extern "C" void kernel_launch(void* const* d_in, const int* in_sizes, int n_in,
                              void* d_out, int out_size, void* d_ws, size_t ws_size,
                              hipStream_t stream)
{
    const float* ue  = (const float*)d_in[0];
    const float* ie  = (const float*)d_in[1];
    const int* tsrc  = (const int*)d_in[2];
    const int* tdst  = (const int*)d_in[3];
    const int* bsrc  = (const int*)d_in[4];
    const int* bdst  = (const int*)d_in[5];

    const long long nU = in_sizes[0] / 64;
    const long long nI = in_sizes[1] / 64;
    const long long N  = nU + nI;
    const int E  = in_sizes[2];
    const int NB = (E > 0) ? (in_sizes[4] / E) : 0;

    const long long Npad = (N + 63) & ~63LL;    // pad rows so k_post tail is safe
    float* ws   = (float*)d_ws;
    float* x0   = ws;                       // Npad*64
    float* het  = x0   + Npad * 64;         // Npad*64
    float* bufA = het  + Npad * 64;         // Npad*64
    float* bufB = bufA + Npad * 64;         // Npad*64
    float* dnS  = bufB + Npad * 64;         // Npad
    float* dnD  = dnS  + Npad;              // Npad
    float* out  = (float*)d_out;

    const int THR = 256;
    const long long tot = N * 64;
    const int ebN = (int)((tot + THR - 1) / THR);

    k_init<<<ebN, THR, 0, stream>>>(ue, ie, x0, out, het, nU * 64, tot);

    auto propagate = [&](const int* src, const int* dst, float* acc) {
        // degree^{-1/2} for src (out) and dst (in)
        hipMemsetAsync(dnS, 0, (size_t)N * sizeof(float), stream);
        hipMemsetAsync(dnD, 0, (size_t)N * sizeof(float), stream);
        int hb = (E + THR - 1) / THR;
        k_hist<<<hb, THR, 0, stream>>>(src, (unsigned*)dnS, E);
        k_hist<<<hb, THR, 0, stream>>>(dst, (unsigned*)dnD, E);
        int nb = (int)((N + THR - 1) / THR);
        k_deg2rnorm<<<nb, THR, 0, stream>>>(dnS, (int)N);
        k_deg2rnorm<<<nb, THR, 0, stream>>>(dnD, (int)N);

        const float* xcur = x0;
        for (int layer = 0; layer < 3; ++layer) {
            float* aggb = (layer & 1) ? bufB : bufA;    // never clobbers xcur
            hipMemsetAsync(aggb, 0, (size_t)N * 64 * sizeof(float), stream);
            long long waves = E;                        // 1 wave per edge
            int wpb = THR / 32;
            int eb  = (int)((waves + wpb - 1) / wpb);
            k_edge<<<eb, THR, 0, stream>>>(src, dst, xcur, dnS, aggb, E);
            int pb = (int)((N + 63) / 64);              // 4 waves x 16 nodes
            k_post<<<pb, 128, 0, stream>>>(aggb, dnD, aggb, acc, (int)N);
            xcur = aggb;
        }
    };

    propagate(tsrc, tdst, out);                 // ui accumulates into d_out
    for (int b = 0; b < NB; ++b)
        propagate(bsrc + (size_t)b * E, bdst + (size_t)b * E, het);

    k_final<<<ebN, THR, 0, stream>>>(out, het, tot);
}